// NucleusSampleFeedBack_43679817400715
// MI455X (gfx1250) — compile-verified
//
#include <hip/hip_runtime.h>
#include <hip/hip_bf16.h>
#include <stdint.h>

// ---------------------------------------------------------------------------
// Nucleus (top-p) sampling, MI455X / gfx1250.
//   grid  = 1024 blocks (one per row), block = 1024 threads (32 wave32 waves)
//   Row (50257 f32 = 201 KB) staged ONCE into 320KB WGP LDS via the Tensor
//   Data Mover; all subsequent passes are LDS-resident.
//   HBM traffic = 206 MB read total -> ~8.8us floor at 23.3 TB/s.
// ---------------------------------------------------------------------------

#define VOCAB     50257
#define NTHREADS  1024
#define NBINS     4096
#define TOPP_F    0.9f
#define TINY_F    1.17549435e-38f

typedef __attribute__((ext_vector_type(4))) unsigned int v4u;
typedef __attribute__((ext_vector_type(8))) int          v8i;
typedef __attribute__((ext_vector_type(4))) int          v4i;

#if __has_builtin(__builtin_amdgcn_tensor_load_to_lds) && \
    __has_builtin(__builtin_amdgcn_s_wait_tensorcnt)
#define USE_TDM 1
#else
#define USE_TDM 0
#endif

// Order-preserving uint32 key for f32 (ascending key == ascending float).
__device__ __forceinline__ uint32_t fkey(float f) {
  uint32_t u = __float_as_uint(f);
  return u ^ ((uint32_t)((int32_t)u >> 31) | 0x80000000u);
}

// JAX-faithful Threefry-2x32 (20 rounds, key injections every 4).
__device__ __forceinline__ void threefry2x32(uint32_t k0, uint32_t k1,
                                             uint32_t x0, uint32_t x1,
                                             uint32_t& o0, uint32_t& o1) {
  const uint32_t ks2 = k0 ^ k1 ^ 0x1BD11BDAu;
  const uint32_t ks[3] = {k0, k1, ks2};
  const int rot[2][4] = {{13, 15, 26, 6}, {17, 29, 16, 24}};
  x0 += k0; x1 += k1;
#pragma unroll
  for (int i = 0; i < 5; ++i) {
#pragma unroll
    for (int r = 0; r < 4; ++r) {
      x0 += x1;
      const int rr = rot[i & 1][r];
      x1 = (x1 << rr) | (x1 >> (32 - rr));
      x1 ^= x0;
    }
    x0 += ks[(i + 1) % 3];
    x1 += ks[(i + 2) % 3] + (uint32_t)(i + 1);
  }
  o0 = x0; o1 = x1;
}

__device__ __forceinline__ float warpRedMax(float v) {
#pragma unroll
  for (int o = 16; o; o >>= 1) v = fmaxf(v, __shfl_xor(v, o, 32));
  return v;
}
__device__ __forceinline__ float warpRedSum(float v) {
#pragma unroll
  for (int o = 16; o; o >>= 1) v += __shfl_xor(v, o, 32);
  return v;
}
__device__ __forceinline__ float warpInclScan(float v) {
  const int lane = threadIdx.x & 31;
#pragma unroll
  for (int o = 1; o < 32; o <<= 1) {
    float n = __shfl_up(v, o, 32);
    if (lane >= o) v += n;
  }
  return v;
}

// Find the bin (scanning bins DESCENDING) where the running mass crosses T.
// Writes the crossing bin index and the mass strictly above it.
__device__ void findCrossing(float T, const float* hist, float* red,
                             unsigned* outBin, float* outAbove) {
  const int tid = threadIdx.x, wid = tid >> 5, lane = tid & 31;
  const int nw = NTHREADS >> 5;
  const int p0 = tid * (NBINS / NTHREADS);           // 4 descending bins/thread
  float h[NBINS / NTHREADS];
  float c = 0.f;
#pragma unroll
  for (int j = 0; j < NBINS / NTHREADS; ++j) {
    h[j] = hist[NBINS - 1 - (p0 + j)];
    c += h[j];
  }
  float incl = warpInclScan(c);
  if (lane == 31) red[wid] = incl;
  __syncthreads();
  if (wid == 0) {
    float wv = (lane < nw) ? red[lane] : 0.f;
    float ws = warpInclScan(wv);
    red[lane] = ws - wv;                              // exclusive wave offsets
  }
  __syncthreads();
  float s = incl - c + red[wid];                      // exclusive prefix (mass above chunk)
#pragma unroll
  for (int j = 0; j < NBINS / NTHREADS; ++j) {
    const float hb = h[j];
    if (s <= T && T < s + hb) {
      *outBin   = (unsigned)(NBINS - 1 - (p0 + j));
      *outAbove = s;
    }
    s += hb;
  }
  __syncthreads();
}

__global__ __launch_bounds__(NTHREADS)
void nucleus_topp_sample_kernel(const float* __restrict__ logits,
                                float* __restrict__ out, int rows) {
  extern __shared__ float row[];                      // VOCAB floats (201 KB)
  __shared__ float    hist[NBINS];                    // 16 KB
  __shared__ float    red[32];
  __shared__ float    fbc[4];                         // 0:max 1:Z 2:above1 3:above2
  __shared__ unsigned ibc[4];                         // 0:b1 1:b2
  __shared__ float    fbest[32];
  __shared__ int      ibest[32];

  const int tid  = threadIdx.x;
  const int wid  = tid >> 5, lane = tid & 31, nw = NTHREADS >> 5;
  const int r    = blockIdx.x;
  if (r >= rows) return;
  const float* __restrict__ grow = logits + (size_t)r * VOCAB;

  // ---- Stage the row into LDS (TDM async DMA; fallback: cooperative copy) --
#if USE_TDM
  if (tid == 0) {
    const uint64_t ga  = (uint64_t)(uintptr_t)grow;
    const uint32_t lds = (uint32_t)(uintptr_t)row;    // low 32b of generic = LDS offset
    v4u g0;
    g0[0] = 1u;                                       // count=1, user mode
    g0[1] = lds;                                      // lds_addr
    g0[2] = (uint32_t)ga;                             // global_addr[31:0]
    g0[3] = (uint32_t)((ga >> 32) & 0x01FFFFFFu) | (2u << 30); // addr[56:32] | type=2
    v8i g1;
    g1[0] = (int)(2u << 16);                          // data_size=4B, mask=0
    g1[1] = (int)(((uint32_t)VOCAB & 0xFFFFu) << 16); // tensor_dim0[15:0]
    g1[2] = (int)((((uint32_t)VOCAB >> 16) & 0xFFFFu) | (1u << 16)); // dim0 hi | tensor_dim1=1
    g1[3] = (int)(((uint32_t)VOCAB & 0xFFFFu) << 16); // tile_dim0 = VOCAB
    g1[4] = 1;                                        // tile_dim1=1, tile_dim2=0
    g1[5] = VOCAB;                                    // tensor_dim0_stride[31:0]
    g1[6] = 0;
    g1[7] = 0;
    v4i g2 = {0, 0, 0, 0};
    v4i g3 = {0, 0, 0, 0};
#if __clang_major__ >= 23
    v8i g4 = {0, 0, 0, 0, 0, 0, 0, 0};
    __builtin_amdgcn_tensor_load_to_lds(g0, g1, g2, g3, g4, 0);
#else
    __builtin_amdgcn_tensor_load_to_lds(g0, g1, g2, g3, 0);
#endif
    __builtin_amdgcn_s_wait_tensorcnt((short)0);
  }
#else
  for (int i = tid; i < VOCAB; i += NTHREADS) row[i] = grow[i];
#endif
  __syncthreads();

  // ---- Pass 1: row max ----------------------------------------------------
  float lm = -INFINITY;
  for (int i = tid; i < VOCAB; i += NTHREADS) lm = fmaxf(lm, row[i]);
  lm = warpRedMax(lm);
  if (lane == 0) red[wid] = lm;
  __syncthreads();
  if (wid == 0) {
    float v = (lane < nw) ? red[lane] : -INFINITY;
    v = warpRedMax(v);
    if (lane == 0) fbc[0] = v;
  }
  __syncthreads();
  const float m = fbc[0];

  // ---- Pass 2: Z = sum exp(x-m) -------------------------------------------
  float ls = 0.f;
  for (int i = tid; i < VOCAB; i += NTHREADS) ls += expf(row[i] - m);
  ls = warpRedSum(ls);
  if (lane == 0) red[wid] = ls;
  __syncthreads();
  if (wid == 0) {
    float v = (lane < nw) ? red[lane] : 0.f;
    v = warpRedSum(v);
    if (lane == 0) fbc[1] = v;
  }
  __syncthreads();
  const float Z = fbc[1];
  const float T = TOPP_F * Z;                         // mass kept (exclusive cumsum <= T)

  // ---- Level-1 histogram over key[31:20] (ds_add_f32 atomics) -------------
  for (int i = tid; i < NBINS; i += NTHREADS) hist[i] = 0.f;
  if (tid == 0) { ibc[0] = 0u; fbc[2] = 0.f; }
  __syncthreads();
  for (int i = tid; i < VOCAB; i += NTHREADS) {
    const float x = row[i];
    atomicAdd(&hist[fkey(x) >> 20], expf(x - m));
  }
  __syncthreads();
  findCrossing(T, hist, red, &ibc[0], &fbc[2]);
  const unsigned b1     = ibc[0];
  const float    above1 = fbc[2];
  const float    T1     = T - above1;

  // ---- Level-2 histogram over key[19:8] within bin b1 ---------------------
  for (int i = tid; i < NBINS; i += NTHREADS) hist[i] = 0.f;
  if (tid == 0) { ibc[1] = 0u; fbc[3] = 0.f; }
  __syncthreads();
  for (int i = tid; i < VOCAB; i += NTHREADS) {
    const float x = row[i];
    const uint32_t k = fkey(x);
    if ((k >> 20) == b1) atomicAdd(&hist[(k >> 8) & 0xFFFu], expf(x - m));
  }
  __syncthreads();
  findCrossing(T1, hist, red, &ibc[1], &fbc[3]);
  const unsigned kcut = (b1 << 12) | ibc[1];          // keep iff (key>>8) >= kcut

  // ---- Sampling: Gumbel-argmax over kept set (UNK idx 0 excluded) ---------
  const uint64_t HALF = ((uint64_t)rows * (uint64_t)VOCAB) >> 1; // JAX counter split
  float best = -INFINITY;
  int   bi   = 0x7fffffff;
  for (int i = tid; i < VOCAB; i += NTHREADS) {
    if (i == 0) continue;                             // word_prob[UNK] = 0
    const float x = row[i];
    const uint32_t k = fkey(x);
    if ((k >> 8) < kcut) continue;                    // masked by top-p
    const uint64_t j = (uint64_t)r * VOCAB + (uint64_t)i;
    uint32_t o0, o1, bits;
    if (j < HALF) { threefry2x32(0u, 1u, (uint32_t)j, (uint32_t)(j + HALF), o0, o1); bits = o0; }
    else          { threefry2x32(0u, 1u, (uint32_t)(j - HALF), (uint32_t)j, o0, o1); bits = o1; }
    const float f = __uint_as_float((bits >> 9) | 0x3f800000u) - 1.0f;
    const float u = fmaxf(f * (1.0f - TINY_F) + TINY_F, TINY_F);
    const float g = -logf(-logf(u));                  // Gumbel(0,1)
    const float sc = x + g;                           // log p + const + gumbel
    if (sc > best || (sc == best && i < bi)) { best = sc; bi = i; }
  }
#pragma unroll
  for (int o = 16; o; o >>= 1) {
    const float ob = __shfl_xor(best, o, 32);
    const int   oi = __shfl_xor(bi, o, 32);
    if (ob > best || (ob == best && oi < bi)) { best = ob; bi = oi; }
  }
  if (lane == 0) { fbest[wid] = best; ibest[wid] = bi; }
  __syncthreads();
  if (tid == 0) {
    float bb = fbest[0]; int bbi = ibest[0];
    for (int w = 1; w < nw; ++w)
      if (fbest[w] > bb || (fbest[w] == bb && ibest[w] < bbi)) { bb = fbest[w]; bbi = ibest[w]; }
    out[r] = (float)bbi;
  }
}

extern "C" void kernel_launch(void* const* d_in, const int* in_sizes, int n_in,
                              void* d_out, int out_size, void* d_ws, size_t ws_size,
                              hipStream_t stream) {
  (void)n_in; (void)d_ws; (void)ws_size; (void)out_size;
  const float* logits = (const float*)d_in[0];
  float* out = (float*)d_out;
  const int rows = in_sizes[0] / VOCAB;               // 1024
  const size_t shmem = (size_t)VOCAB * sizeof(float); // 201,028 B dynamic LDS
  hipLaunchKernelGGL(nucleus_topp_sample_kernel, dim3(rows), dim3(NTHREADS),
                     shmem, stream, logits, out, rows);
}